// GC_3D_v2_31568009625984
// MI455X (gfx1250) — compile-verified
//
#include <hip/hip_runtime.h>

// ---------------- problem geometry ----------------
#define BATCH 4
#define ZDIM 128
#define YDIM 192
#define XDIM 192
#define TX 64                 // x tile (owned)
#define TY 16                 // y tile (owned)
#define KY 4                  // rows owned per thread (register blocking)
#define ZS 32                 // z planes per slab
#define XT (XDIM / TX)        // 3
#define YT (YDIM / TY)        // 12
#define ZT (ZDIM / ZS)        // 4
#define NWG (XT * YT * ZT * BATCH)   // 576
#define HX (TX + 2)           // 66  (halo tile width)
#define HY (TY + 2)           // 18  (halo tile height)
#define PLANE (HX * HY)       // 1188 floats per plane tile
#define NSLOT 5               // ceil(PLANE / 256)
#define NACC 27               // 13 didt + 13 dt2 + 1 bce
#define NTOT ((double)BATCH * ZDIM * YDIM * XDIM)

#define GC_USE_ASYNC 1        // CDNA5 GLOBAL_LOAD_ASYNC_TO_LDS pipeline

typedef float v2f __attribute__((ext_vector_type(2)));
typedef float v8f __attribute__((ext_vector_type(8)));

// ---------------- async helpers ----------------
__device__ __forceinline__ void wait_async0() {
#if __has_builtin(__builtin_amdgcn_s_wait_asynccnt)
  __builtin_amdgcn_s_wait_asynccnt(0);
#else
  asm volatile("s_wait_asynccnt 0" ::: "memory");
#endif
}

// Per ISA 10.2: flat address of an LDS pointer carries the wave-relative LDS
// byte offset in its low 32 bits -> that is exactly the VDST operand of
// GLOBAL_LOAD_ASYNC_TO_LDS.
__device__ __forceinline__ unsigned lds_off32(const float* p) {
  return (unsigned)(uintptr_t)p;
}

__device__ __forceinline__ void async_copy_b32(const float* g, float* lds) {
#if GC_USE_ASYNC
  asm volatile("global_load_async_to_lds_b32 %0, %1, off"
               :: "v"(lds_off32(lds)), "v"(g)
               : "memory");
#else
  *lds = *g;
#endif
}

// ---------------- wave32 reduction via V_WMMA_F32_16X16X4_F32 ----------------
// A: vgpr0 = per-lane value (A[m][0]=v[m], A[m][2]=v[m+16]), vgpr1 = 0.
// B: all ones.  D[m][n] = v[m] + v[m+16].  Sum the 8 D vgprs, then one
// xor-16 shuffle completes the deterministic 32-lane sum on the matrix pipe.
__device__ __forceinline__ float wave_sum_wmma(float v) {
  v2f a; a[0] = v;    a[1] = 0.0f;
  v2f b; b[0] = 1.0f; b[1] = 1.0f;
  v8f c = {};
  v8f d = __builtin_amdgcn_wmma_f32_16x16x4_f32(false, a, false, b,
                                                (short)0, c, false, false);
  float s = d[0] + d[1] + d[2] + d[3] + d[4] + d[5] + d[6] + d[7];
  s += __shfl_xor(s, 16, 32);
  return s;
}

// ---------------- per-voxel accumulation ----------------
// Canonical offsets (dz,dy,dx), dz in {0,1}; b = a - o.
// j:  0:(1,0,0) 1:(0,1,0) 2:(0,0,1) 3:(1,1,0) 4:(1,-1,0) 5:(0,1,1) 6:(0,1,-1)
//     7:(1,0,-1) 8:(1,0,1) 9:(1,-1,-1) 10:(1,1,-1) 11:(1,1,1) 12:(1,-1,1)
template <bool CHK>
__device__ __forceinline__ void accum_voxel(
    const float* __restrict__ cin, const float* __restrict__ ctg,
    const float* __restrict__ pin, const float* __restrict__ ptg,
    int ly, int lx, int gy, int gx, int z,
    float* didt, float* dt2, float& bce)
{
  const int idx = ly * HX + lx;
  const float ci = cin[idx];
  const float ct = ctg[idx];

  bce += ct * __logf(ci) + (1.0f - ct) * __logf(1.0f - ci);

  float w0 = 1.0f, w1 = 1.0f, w2 = 1.0f, w3 = 1.0f, w4 = 1.0f, w5 = 1.0f,
        w6 = 1.0f, w7 = 1.0f, w8 = 1.0f, w9 = 1.0f, w10 = 1.0f, w11 = 1.0f,
        w12 = 1.0f;
  if (CHK) {
    const float wz  = (z  >= 1)        ? 1.0f : 0.0f;
    const float wyl = (gy >= 1)        ? 1.0f : 0.0f;
    const float wyh = (gy <= YDIM - 2) ? 1.0f : 0.0f;
    const float wxl = (gx >= 1)        ? 1.0f : 0.0f;
    const float wxh = (gx <= XDIM - 2) ? 1.0f : 0.0f;
    w0 = wz;        w1 = wyl;       w2 = wxl;
    w3 = wz * wyl;  w4 = wz * wyh;
    w5 = wyl * wxl; w6 = wyl * wxh;
    w7 = wz * wxh;  w8 = wz * wxl;
    w9 = w4 * wxh;  w10 = w3 * wxh; w11 = w3 * wxl; w12 = w4 * wxl;
  }

#define GC_ACC(J, BI, BT, D, W)                                   \
  do {                                                            \
    float ni = BI[idx + (D)];                                     \
    float nt = BT[idx + (D)];                                     \
    float di = ci - ni;                                           \
    float dt = ct - nt;                                           \
    float dtw = CHK ? (W) * dt : dt;                              \
    didt[J] += di * dtw;                                          \
    dt2[J]  += dt * dtw;                                          \
  } while (0)

  GC_ACC(0,  pin, ptg,  0,      w0);   // (1,0,0)
  GC_ACC(1,  cin, ctg, -HX,     w1);   // (0,1,0)
  GC_ACC(2,  cin, ctg, -1,      w2);   // (0,0,1)
  GC_ACC(3,  pin, ptg, -HX,     w3);   // (1,1,0)
  GC_ACC(4,  pin, ptg,  HX,     w4);   // (1,-1,0)
  GC_ACC(5,  cin, ctg, -HX - 1, w5);   // (0,1,1)
  GC_ACC(6,  cin, ctg, -HX + 1, w6);   // (0,1,-1)
  GC_ACC(7,  pin, ptg,  1,      w7);   // (1,0,-1)
  GC_ACC(8,  pin, ptg, -1,      w8);   // (1,0,1)
  GC_ACC(9,  pin, ptg,  HX + 1, w9);   // (1,-1,-1)
  GC_ACC(10, pin, ptg, -HX + 1, w10);  // (1,1,-1)
  GC_ACC(11, pin, ptg, -HX - 1, w11);  // (1,1,1)  [== (-1,-1,-1)]
  GC_ACC(12, pin, ptg,  HX - 1, w12);  // (1,-1,1) [== (-1,1,-1)]
#undef GC_ACC
}

// ---------------- main streaming kernel ----------------
__global__ void __launch_bounds__(256) gc3d_main(const float* __restrict__ gin,
                                                 const float* __restrict__ gtg,
                                                 float* __restrict__ ws)
{
  __shared__ float sIn[3][PLANE];   // triple-buffered plane tiles
  __shared__ float sTg[3][PLANE];
  __shared__ float sBins[8][NACC];

  const int tid   = threadIdx.x;
  const int bx    = blockIdx.x;   // 0..35 : xy tile
  const int zslab = blockIdx.y;   // 0..3
  const int batch = blockIdx.z;   // 0..3
  const int xt = bx % XT;
  const int yt = bx / XT;
  const int gx0 = xt * TX;
  const int gy0 = yt * TY;
  const int z0 = zslab * ZS;
  const int z1 = z0 + ZS;

  const int tx  = tid & (TX - 1);   // 0..63
  const int tyg = tid >> 6;         // 0..3
  const int gx  = gx0 + tx;
  const int gyb = gy0 + tyg * KY;   // first owned row
  const int lx  = 1 + tx;
  const int lyb = 1 + tyg * KY;

  const size_t pstride = (size_t)YDIM * XDIM;
  const float* inBase = gin + (size_t)batch * ZDIM * pstride;
  const float* tgBase = gtg + (size_t)batch * ZDIM * pstride;

  // ---- precompute this thread's halo-tile load slots (fixed across z) ----
  int  goff[NSLOT];
  int  loff[NSLOT];
  bool lok[NSLOT];
#pragma unroll
  for (int s = 0; s < NSLOT; ++s) {
    const int l  = tid + s * 256;
    const int yy = l / HX;
    const int xx = l - yy * HX;
    const int gy = gy0 - 1 + yy;
    const int gg = gx0 - 1 + xx;
    lok[s]  = (l < PLANE) && (gy >= 0) && (gy < YDIM) && (gg >= 0) && (gg < XDIM);
    goff[s] = gy * XDIM + gg;
    loff[s] = l;
  }

  // ---- zero all buffers once: OOB halo slots stay 0 for the whole slab ----
  {
    float* zi = &sIn[0][0];
    float* zt = &sTg[0][0];
    for (int l = tid; l < 3 * PLANE; l += 256) { zi[l] = 0.0f; zt[l] = 0.0f; }
  }
  __syncthreads();   // LDS zeros visible before the DMA engine writes

#define GC_ISSUE_PLANE(BASE, LBUF)                                 \
  do {                                                             \
    _Pragma("unroll")                                              \
    for (int s = 0; s < NSLOT; ++s)                                \
      if (lok[s]) async_copy_b32((BASE) + goff[s], (LBUF) + loff[s]); \
  } while (0)

  // ---- prime the pipeline: prev -> (z0+2)%3, cur -> z0%3 ----
  int ic = z0 % 3;          // cur buffer
  int ip = (z0 + 2) % 3;    // prev buffer (stays zero when z0 == 0)
  if (z0 > 0) {
    GC_ISSUE_PLANE(inBase + (size_t)(z0 - 1) * pstride, sIn[ip]);
    GC_ISSUE_PLANE(tgBase + (size_t)(z0 - 1) * pstride, sTg[ip]);
  }
  GC_ISSUE_PLANE(inBase + (size_t)z0 * pstride, sIn[ic]);
  GC_ISSUE_PLANE(tgBase + (size_t)z0 * pstride, sTg[ic]);
  wait_async0();
  __syncthreads();

  float didt[13], dt2[13];
  float bce = 0.0f;
#pragma unroll
  for (int j = 0; j < 13; ++j) { didt[j] = 0.0f; dt2[j] = 0.0f; }

  const bool fast_xy = (gx >= 1) && (gx <= XDIM - 2) &&
                       (gyb >= 1) && (gyb + KY - 1 <= YDIM - 2);

  for (int z = z0; z < z1; ++z) {
    const int inx = 3 - ic - ip;   // third buffer = next plane

    // fire async DMA for plane z+1 while we compute plane z
    if (z + 1 < z1) {
      GC_ISSUE_PLANE(inBase + (size_t)(z + 1) * pstride, sIn[inx]);
      GC_ISSUE_PLANE(tgBase + (size_t)(z + 1) * pstride, sTg[inx]);
    }
    if (z + 2 < z1) {   // nudge z+2 toward L2 for the next DMA round
      __builtin_prefetch(inBase + (size_t)(z + 2) * pstride + gyb * XDIM + gx, 0, 1);
      __builtin_prefetch(tgBase + (size_t)(z + 2) * pstride + gyb * XDIM + gx, 0, 1);
    }

    const float* cin = sIn[ic];
    const float* ctg = sTg[ic];
    const float* pin = sIn[ip];
    const float* ptg = sTg[ip];

    if (fast_xy && z >= 1) {
#pragma unroll
      for (int r = 0; r < KY; ++r)
        accum_voxel<false>(cin, ctg, pin, ptg, lyb + r, lx, gyb + r, gx, z,
                           didt, dt2, bce);
    } else {
#pragma unroll
      for (int r = 0; r < KY; ++r)
        accum_voxel<true>(cin, ctg, pin, ptg, lyb + r, lx, gyb + r, gx, z,
                          didt, dt2, bce);
    }

    wait_async0();     // our async writes to LDS have landed
    __syncthreads();   // everyone's writes visible; prev buffer reusable
    ip = ic;
    ic = inx;
  }
#undef GC_ISSUE_PLANE

  // ---- deterministic workgroup reduction: WMMA wave sums -> LDS -> ws ----
  const int lane = tid & 31;
  const int wave = tid >> 5;
  {
    float t2[NACC];
#pragma unroll
    for (int j = 0; j < 13; ++j) t2[j] = wave_sum_wmma(didt[j]);
#pragma unroll
    for (int j = 0; j < 13; ++j) t2[13 + j] = wave_sum_wmma(dt2[j]);
    t2[26] = wave_sum_wmma(bce);
    if (lane == 0) {
#pragma unroll
      for (int j = 0; j < NACC; ++j) sBins[wave][j] = t2[j];
    }
  }
  __syncthreads();
  if (tid < NACC) {
    float s = 0.0f;
#pragma unroll
    for (int w = 0; w < 8; ++w) s += sBins[w][tid];   // fixed order
    const int wg = (batch * ZT + zslab) * (XT * YT) + bx;
    ws[(size_t)wg * NACC + tid] = s;
  }
}

// ---------------- finalize: deterministic tree over 576 partials ----------------
__global__ void __launch_bounds__(256) gc3d_final(const float* __restrict__ ws,
                                                  float* __restrict__ out)
{
  __shared__ float part[8][NACC];
  __shared__ float fin[NACC];
  const int tid  = threadIdx.x;
  const int lane = tid & 31;
  const int wave = tid >> 5;
  const int per  = NWG / 8;  // 72 workgroup-partials per wave

  if (lane < NACC) {
    float s = 0.0f;
    const float* p = ws + (size_t)wave * per * NACC + lane;
    for (int i = 0; i < per; ++i) s += p[(size_t)i * NACC];  // fixed order
    part[wave][lane] = s;
  }
  __syncthreads();
  if (tid < NACC) {
    float s = 0.0f;
#pragma unroll
    for (int w = 0; w < 8; ++w) s += part[w][tid];           // fixed order
    fin[tid] = s;
  }
  __syncthreads();
  if (tid == 0) {
    float acc = 0.0f;
#pragma unroll
    for (int j = 0; j < 13; ++j)
      acc += fin[j] / (fin[13 + j] + 1e-5f);
    const float bce = (float)(-(double)fin[26] / NTOT);
    out[0] = bce + (1.0f - acc / 13.0f);
  }
}

// ---------------- entry point ----------------
extern "C" void kernel_launch(void* const* d_in, const int* in_sizes, int n_in,
                              void* d_out, int out_size, void* d_ws, size_t ws_size,
                              hipStream_t stream) {
  (void)in_sizes; (void)n_in; (void)out_size; (void)ws_size;
  const float* gin = (const float*)d_in[0];   // "input"
  const float* gtg = (const float*)d_in[1];   // "target"
  float* out = (float*)d_out;
  float* ws  = (float*)d_ws;                  // NWG * NACC floats (~62 KB)

  dim3 grid(XT * YT, ZT, BATCH);
  gc3d_main<<<grid, 256, 0, stream>>>(gin, gtg, ws);
  gc3d_final<<<1, 256, 0, stream>>>(ws, out);
}